// GATMultiHead_residual_2894807957586
// MI455X (gfx1250) — compile-verified
//
#include <hip/hip_runtime.h>
#include <hip/hip_bf16.h>
#include <math.h>

#define NNODES 100000
#define NEDGES 1600000
#define NGRAPH 256
#define NHEAD  2

typedef float v2f __attribute__((ext_vector_type(2)));
typedef float v8f __attribute__((ext_vector_type(8)));

// ---------------------------------------------------------------------------
// helpers
// ---------------------------------------------------------------------------
__device__ __forceinline__ float atomicMaxFloat(float* addr, float value) {
  // sign-split trick: positive floats order like ints, negative like reversed uints
  if (value >= 0.0f)
    return __int_as_float(atomicMax((int*)addr, __float_as_int(value)));
  else
    return __uint_as_float(atomicMin((unsigned int*)addr, __float_as_uint(value)));
}

// ---------------------------------------------------------------------------
// pad x[N,5] -> xp[N,8] (zero-filled) so all GEMM K dims are multiples of 4
// ---------------------------------------------------------------------------
__global__ void pad_x_kernel(const float* __restrict__ x, float* __restrict__ xp, int Nn) {
  int i = blockIdx.x * blockDim.x + threadIdx.x;
  if (i >= Nn * 8) return;
  int n = i >> 3, c = i & 7;
  xp[i] = (c < 5) ? x[n * 5 + c] : 0.0f;
}

// ---------------------------------------------------------------------------
// f32 WMMA GEMM:  C[M x NCOL] = A[M x K] @ B[K x NCOL] (+ bias[NCOL])
// one wave32 per 16x16 tile, V_WMMA_F32_16X16X4_F32, K-step 4, no divergence.
// K, NCOL compile-time; M multiple of 16 (M = 100000 = 6250*16).
// ---------------------------------------------------------------------------
template <int K, int NCOL>
__global__ void wmma_gemm_f32(const float* __restrict__ A,
                              const float* __restrict__ B,
                              const float* __restrict__ bias,
                              float* __restrict__ C, int M) {
  const int lane   = threadIdx.x & 31;
  const int wave   = (blockIdx.x * blockDim.x + threadIdx.x) >> 5;  // wave-uniform
  const int tilesN = NCOL >> 4;
  if (wave >= (M >> 4) * tilesN) return;          // whole-wave uniform exit
  const int tm = wave / tilesN;
  const int tn = wave - tm * tilesN;
  const int m0 = tm << 4;
  const int n0 = tn << 4;

  const int half = lane >> 4;     // 0: lanes 0-15 (K pair 0,1), 1: lanes 16-31 (K pair 2,3)
  const int l    = lane & 15;

  // A 16x4 layout: lane row = m0+l, VGPR0/1 hold K = k+2*half, k+2*half+1
  const v2f*   __restrict__ arow = (const v2f*)(A + (size_t)(m0 + l) * K);
  const float* __restrict__ bcol = B + n0 + l;

  v8f acc = {};
#pragma unroll 4
  for (int k = 0; k < K; k += 4) {
    v2f a = arow[(k >> 1) + half];              // global_load_b64
    const int k0 = k + 2 * half;
    v2f b;
    b.x = bcol[(size_t)k0 * NCOL];
    b.y = bcol[(size_t)(k0 + 1) * NCOL];
    acc = __builtin_amdgcn_wmma_f32_16x16x4_f32(false, a, false, b,
                                                (short)0, acc, false, false);
  }

  // D 16x16 f32: VGPR r -> row m0+r (lanes 0-15) / m0+r+8 (lanes 16-31)
  const int   col  = n0 + l;
  const float badd = bias ? bias[col] : 0.0f;
  const int   rb   = m0 + 8 * half;
#pragma unroll
  for (int r = 0; r < 8; ++r)
    C[(size_t)(rb + r) * NCOL + col] = acc[r] + badd;
}

// ---------------------------------------------------------------------------
// generic fill
// ---------------------------------------------------------------------------
__global__ void fill_f32(float* __restrict__ p, float v, int n) {
  int i = blockIdx.x * blockDim.x + threadIdx.x;
  if (i < n) p[i] = v;
}

// ---------------------------------------------------------------------------
// attention logits: e_src[n,h] = dot(hfeat[n,h,:], a_src[h,:]); same for dst
// ---------------------------------------------------------------------------
__global__ void edot_kernel(const float* __restrict__ hfeat,
                            const float* __restrict__ a_src,
                            const float* __restrict__ a_dst,
                            float* __restrict__ e_src,
                            float* __restrict__ e_dst,
                            int Nn, int Hh, int C) {
  int i = blockIdx.x * blockDim.x + threadIdx.x;
  if (i >= Nn * Hh) return;
  int hd = i % Hh;
  const float* hp = hfeat + (size_t)i * C;
  const float* as = a_src + hd * C;
  const float* ad = a_dst + hd * C;
  float s1 = 0.f, s2 = 0.f;
  for (int c = 0; c < C; c += 4) {
    float4 v = *(const float4*)(hp + c);
    float4 a = *(const float4*)(as + c);
    float4 b = *(const float4*)(ad + c);
    s1 += v.x * a.x + v.y * a.y + v.z * a.z + v.w * a.w;
    s2 += v.x * b.x + v.y * b.y + v.z * b.z + v.w * b.w;
  }
  e_src[i] = s1;
  e_dst[i] = s2;
}

// ---------------------------------------------------------------------------
// edge pass 1: segment max over dst of leaky_relu(e_src[src]+e_dst[dst])
// edges i<E from edge_index; i>=E are the N self loops
// ---------------------------------------------------------------------------
__global__ void edge_max_kernel(const int* __restrict__ ei, int E, int Nn,
                                const float* __restrict__ e_src,
                                const float* __restrict__ e_dst,
                                float* __restrict__ amax, int Hh) {
  int i = blockIdx.x * blockDim.x + threadIdx.x;
  int Et = E + Nn;
  if (i >= Et) return;
  int s, d;
  if (i < E) { s = ei[i]; d = ei[E + i]; } else { s = d = i - E; }
  for (int hd = 0; hd < Hh; ++hd) {
    float a = e_src[s * Hh + hd] + e_dst[d * Hh + hd];
    a = (a > 0.f) ? a : 0.2f * a;            // leaky_relu(0.2)
    atomicMaxFloat(&amax[d * Hh + hd], a);
  }
}

// ---------------------------------------------------------------------------
// edge pass 2: ex = exp(alpha - amax[dst]); denom += ex; acc[dst] += h[src]*ex
// thread = (edge, head, 4-channel group) -> global_load_b128 gather of h[src]
// ---------------------------------------------------------------------------
__global__ void edge_agg_kernel(const int* __restrict__ ei, int E, int Nn,
                                const float* __restrict__ e_src,
                                const float* __restrict__ e_dst,
                                const float* __restrict__ amax,
                                const float* __restrict__ hfeat,
                                float* __restrict__ denom,
                                float* __restrict__ acc,
                                int Hh, int C) {
  int i = blockIdx.x * blockDim.x + threadIdx.x;
  const int C4 = C >> 2;
  const int Et = E + Nn;
  const int total = Et * Hh * C4;              // <= 54.4M, fits int
  if (i >= total) return;
  int c4 = i % C4;
  int t  = i / C4;
  int hd = t % Hh;
  int e  = t / Hh;
  int s, d;
  if (e < E) { s = ei[e]; d = ei[E + e]; } else { s = d = e - E; }
  float a = e_src[s * Hh + hd] + e_dst[d * Hh + hd];
  a = (a > 0.f) ? a : 0.2f * a;
  float ex = expf(a - amax[d * Hh + hd]);
  if (c4 == 0) atomicAdd(&denom[d * Hh + hd], ex);
  float4 hv = *(const float4*)(hfeat + ((size_t)s * Hh + hd) * C + 4 * c4);
  float* ap = acc + ((size_t)d * Hh + hd) * C + 4 * c4;
  atomicAdd(ap + 0, hv.x * ex);
  atomicAdd(ap + 1, hv.y * ex);
  atomicAdd(ap + 2, hv.z * ex);
  atomicAdd(ap + 3, hv.w * ex);
}

// ---------------------------------------------------------------------------
// node epilogue: y = relu(BN(acc/denom + bias)) + y   (y pre-holds projection)
// ---------------------------------------------------------------------------
__global__ void epilogue_kernel(const float* __restrict__ acc,
                                const float* __restrict__ denom,
                                const float* __restrict__ bias,
                                const float* __restrict__ bg,
                                const float* __restrict__ bb,
                                const float* __restrict__ bm,
                                const float* __restrict__ bv,
                                float* __restrict__ y,
                                int Nn, int Hh, int C) {
  int HC = Hh * C;
  int i = blockIdx.x * blockDim.x + threadIdx.x;
  if (i >= Nn * HC) return;
  int n = i / HC;
  int j = i - n * HC;
  int hd = j / C;
  float x = acc[i] / (denom[n * Hh + hd] + 1e-16f) + bias[j];
  x = bg[j] * (x - bm[j]) * rsqrtf(bv[j] + 1e-5f) + bb[j];
  x = (x > 0.f) ? x : 0.f;
  y[i] = x + y[i];
}

// ---------------------------------------------------------------------------
// per-graph mean pooling (sum + count via atomics; divide in head kernel)
// thread = (node, 4-channel group)
// ---------------------------------------------------------------------------
__global__ void pool_kernel(const float* __restrict__ hfeat,
                            const int* __restrict__ batch,
                            float* __restrict__ pooled,
                            float* __restrict__ counts,
                            int Nn, int F) {
  int i = blockIdx.x * blockDim.x + threadIdx.x;
  const int F4 = F >> 2;
  if (i >= Nn * F4) return;
  int n  = i / F4;
  int c4 = i - n * F4;
  int g  = batch[n];
  float4 hv = *(const float4*)(hfeat + (size_t)n * F + 4 * c4);
  float* pp = pooled + (size_t)g * F + 4 * c4;
  atomicAdd(pp + 0, hv.x);
  atomicAdd(pp + 1, hv.y);
  atomicAdd(pp + 2, hv.z);
  atomicAdd(pp + 3, hv.w);
  if (c4 == 0) atomicAdd(&counts[g], 1.0f);
}

// ---------------------------------------------------------------------------
// MLP head: z = relu(BN(pooled@fw+fb)); z = relu(z@l1w+l1b); out = z@l2w+l2b
// one block (32 threads) per graph
// ---------------------------------------------------------------------------
__global__ void head_kernel(const float* __restrict__ pooled,
                            const float* __restrict__ counts,
                            const float* __restrict__ fw, const float* __restrict__ fb,
                            const float* __restrict__ g4, const float* __restrict__ be4,
                            const float* __restrict__ m4, const float* __restrict__ v4,
                            const float* __restrict__ l1w, const float* __restrict__ l1b,
                            const float* __restrict__ l2w, const float* __restrict__ l2b,
                            float* __restrict__ out) {
  int g = blockIdx.x;
  int t = threadIdx.x;                       // 0..31
  __shared__ float p[128];
  __shared__ float z1[32];
  __shared__ float z2[32];
  float cnt = counts[g];
  cnt = (cnt > 1.0f) ? cnt : 1.0f;
  for (int k = t; k < 128; k += 32) p[k] = pooled[g * 128 + k] / cnt;
  __syncthreads();
  float s = fb[t];
  for (int k = 0; k < 128; ++k) s += p[k] * fw[k * 32 + t];
  s = g4[t] * (s - m4[t]) * rsqrtf(v4[t] + 1e-5f) + be4[t];
  z1[t] = (s > 0.f) ? s : 0.f;
  __syncthreads();
  float s2 = l1b[t];
  for (int k = 0; k < 32; ++k) s2 += z1[k] * l1w[k * 32 + t];
  z2[t] = (s2 > 0.f) ? s2 : 0.f;
  __syncthreads();
  if (t < 10) {
    float s3 = l2b[t];
    for (int k = 0; k < 32; ++k) s3 += z2[k] * l2w[k * 10 + t];
    out[g * 10 + t] = s3;
  }
}

// ---------------------------------------------------------------------------
// host orchestration
// ---------------------------------------------------------------------------
static inline int ceil_div(int a, int b) { return (a + b - 1) / b; }

// attention + softmax-aggregate + BN/relu/residual epilogue (yout pre-holds proj)
static void run_attention(int C,
                          const float* aS, const float* aD, const float* bias,
                          const float* bg, const float* bb, const float* bm, const float* bv,
                          float* yout, float* hbuf, float* accbuf,
                          float* e_src, float* e_dst, float* amax, float* denom,
                          const int* ei, hipStream_t stream) {
  const int HC = NHEAD * C;
  edot_kernel<<<ceil_div(NNODES * NHEAD, 256), 256, 0, stream>>>(hbuf, aS, aD, e_src, e_dst,
                                                                 NNODES, NHEAD, C);
  fill_f32<<<ceil_div(NNODES * NHEAD, 256), 256, 0, stream>>>(amax, -INFINITY, NNODES * NHEAD);
  fill_f32<<<ceil_div(NNODES * NHEAD, 256), 256, 0, stream>>>(denom, 0.0f, NNODES * NHEAD);
  fill_f32<<<ceil_div(NNODES * HC, 256), 256, 0, stream>>>(accbuf, 0.0f, NNODES * HC);
  const int Et = NEDGES + NNODES;
  edge_max_kernel<<<ceil_div(Et, 256), 256, 0, stream>>>(ei, NEDGES, NNODES,
                                                         e_src, e_dst, amax, NHEAD);
  edge_agg_kernel<<<ceil_div(Et * NHEAD * (C >> 2), 256), 256, 0, stream>>>(
      ei, NEDGES, NNODES, e_src, e_dst, amax, hbuf, denom, accbuf, NHEAD, C);
  epilogue_kernel<<<ceil_div(NNODES * HC, 256), 256, 0, stream>>>(
      accbuf, denom, bias, bg, bb, bm, bv, yout, NNODES, NHEAD, C);
}

extern "C" void kernel_launch(void* const* d_in, const int* in_sizes, int n_in,
                              void* d_out, int out_size, void* d_ws, size_t ws_size,
                              hipStream_t stream) {
  // inputs in setup_inputs() dict order
  const float* x    = (const float*)d_in[0];
  const int*   ei   = (const int*)d_in[1];
  const int*   bat  = (const int*)d_in[2];
  const float* w1   = (const float*)d_in[3];
  const float* as1  = (const float*)d_in[4];
  const float* ad1  = (const float*)d_in[5];
  const float* b1   = (const float*)d_in[6];
  const float* g1   = (const float*)d_in[7];
  const float* be1  = (const float*)d_in[8];
  const float* m1   = (const float*)d_in[9];
  const float* v1   = (const float*)d_in[10];
  const float* p1w  = (const float*)d_in[11];
  const float* p1b  = (const float*)d_in[12];
  const float* w2   = (const float*)d_in[13];
  const float* as2  = (const float*)d_in[14];
  const float* ad2  = (const float*)d_in[15];
  const float* b2   = (const float*)d_in[16];
  const float* g2   = (const float*)d_in[17];
  const float* be2  = (const float*)d_in[18];
  const float* m2   = (const float*)d_in[19];
  const float* v2   = (const float*)d_in[20];
  const float* p2w  = (const float*)d_in[21];
  const float* p2b  = (const float*)d_in[22];
  const float* w3   = (const float*)d_in[23];
  const float* as3  = (const float*)d_in[24];
  const float* ad3  = (const float*)d_in[25];
  const float* b3   = (const float*)d_in[26];
  const float* g3   = (const float*)d_in[27];
  const float* be3  = (const float*)d_in[28];
  const float* m3   = (const float*)d_in[29];
  const float* v3   = (const float*)d_in[30];
  const float* p3w  = (const float*)d_in[31];
  const float* p3b  = (const float*)d_in[32];
  const float* fw   = (const float*)d_in[33];
  const float* fb   = (const float*)d_in[34];
  const float* g4   = (const float*)d_in[35];
  const float* be4  = (const float*)d_in[36];
  const float* m4   = (const float*)d_in[37];
  const float* v4   = (const float*)d_in[38];
  const float* l1w  = (const float*)d_in[39];
  const float* l1b  = (const float*)d_in[40];
  const float* l2w  = (const float*)d_in[41];
  const float* l2b  = (const float*)d_in[42];
  float* out = (float*)d_out;

  // workspace layout (floats)
  float* ws = (float*)d_ws;
  const size_t BIG = (size_t)NNODES * 128;
  float* hbuf   = ws;                 // [N,128] transformed features
  float* accbuf = ws + BIG;           // [N,128] segment-sum accumulator
  float* bufA   = ws + 2 * BIG;       // [N,128] ping
  float* bufB   = ws + 3 * BIG;       // [N,128] pong
  float* e_src  = ws + 4 * BIG;                         // [N,2]
  float* e_dst  = e_src + (size_t)NNODES * NHEAD;       // [N,2]
  float* amax   = e_dst + (size_t)NNODES * NHEAD;       // [N,2]
  float* denom  = amax  + (size_t)NNODES * NHEAD;       // [N,2]
  float* pooled = denom + (size_t)NNODES * NHEAD;       // [G,128]
  float* counts = pooled + (size_t)NGRAPH * 128;        // [G]
  float* xpad   = counts + NGRAPH;                      // [N,8]

  const int WPB = 256 / 32;  // waves per 256-thread block

  // pad x to K=8
  pad_x_kernel<<<ceil_div(NNODES * 8, 256), 256, 0, stream>>>(x, xpad, NNODES);

  // ---- layer 1: x[N,8] -> bufA[N,64], C=32 ----
  {
    const int tiles = (NNODES / 16) * (64 / 16);
    wmma_gemm_f32<8, 64><<<ceil_div(tiles, WPB), 256, 0, stream>>>(xpad, w1, nullptr, hbuf, NNODES);
    wmma_gemm_f32<8, 64><<<ceil_div(tiles, WPB), 256, 0, stream>>>(xpad, p1w, p1b, bufA, NNODES);
    run_attention(32, as1, ad1, b1, g1, be1, m1, v1, bufA, hbuf, accbuf,
                  e_src, e_dst, amax, denom, ei, stream);
  }
  // ---- layer 2: bufA[N,64] -> bufB[N,128], C=64 ----
  {
    const int tiles = (NNODES / 16) * (128 / 16);
    wmma_gemm_f32<64, 128><<<ceil_div(tiles, WPB), 256, 0, stream>>>(bufA, w2, nullptr, hbuf, NNODES);
    wmma_gemm_f32<64, 128><<<ceil_div(tiles, WPB), 256, 0, stream>>>(bufA, p2w, p2b, bufB, NNODES);
    run_attention(64, as2, ad2, b2, g2, be2, m2, v2, bufB, hbuf, accbuf,
                  e_src, e_dst, amax, denom, ei, stream);
  }
  // ---- layer 3: bufB[N,128] -> bufA[N,128], C=64 ----
  {
    const int tiles = (NNODES / 16) * (128 / 16);
    wmma_gemm_f32<128, 128><<<ceil_div(tiles, WPB), 256, 0, stream>>>(bufB, w3, nullptr, hbuf, NNODES);
    wmma_gemm_f32<128, 128><<<ceil_div(tiles, WPB), 256, 0, stream>>>(bufB, p3w, p3b, bufA, NNODES);
    run_attention(64, as3, ad3, b3, g3, be3, m3, v3, bufA, hbuf, accbuf,
                  e_src, e_dst, amax, denom, ei, stream);
  }

  // ---- mean pool per graph + MLP head ----
  fill_f32<<<ceil_div(NGRAPH * 128, 256), 256, 0, stream>>>(pooled, 0.0f, NGRAPH * 128);
  fill_f32<<<ceil_div(NGRAPH, 256), 256, 0, stream>>>(counts, 0.0f, NGRAPH);
  pool_kernel<<<ceil_div(NNODES * 32, 256), 256, 0, stream>>>(bufA, bat, pooled, counts,
                                                              NNODES, 128);
  head_kernel<<<NGRAPH, 32, 0, stream>>>(pooled, counts, fw, fb, g4, be4, m4, v4,
                                         l1w, l1b, l2w, l2b, out);
}